// GCNNet_56959856279864
// MI455X (gfx1250) — compile-verified
//
#include <hip/hip_runtime.h>

// ---------------------------------------------------------------------------
// GCN 2-layer forward for MI455X (gfx1250, wave32).
//  x:[N,128] f32, W1:[128,32], b1:[32], W2:[32,16], b2:[16], edge_index:[2,E] i32
//  out:[N,16] f32
// Dense transforms use V_WMMA_F32_16X16X4_F32 (fp32 matrix pipe).
// Edge aggregation uses hardware fp32 atomics (L2-resident destinations).
// ---------------------------------------------------------------------------

#define N_NODES 100000
#define N_EDGES 3200000
#define IN_DIM  128
#define HID_DIM 32
#define OUT_DIM 16

typedef __attribute__((ext_vector_type(2))) float v2f;
typedef __attribute__((ext_vector_type(8))) float v8f;

// ---------------- init: deg accumulator = 1 (self loop), agg1 = 0, out = b2
__global__ void __launch_bounds__(256) k_init(float* __restrict__ deg,
                                              float* __restrict__ agg1,
                                              float* __restrict__ out,
                                              const float* __restrict__ b2) {
  int i = blockIdx.x * blockDim.x + threadIdx.x;
  if (i < N_NODES) deg[i] = 1.0f;
  if (i < N_NODES * HID_DIM) agg1[i] = 0.0f;
  if (i < N_NODES * OUT_DIM) out[i] = b2[i & (OUT_DIM - 1)];
}

// ---------------- in-degree via fp32 atomics (deg[dst] += 1)
__global__ void __launch_bounds__(256) k_degree(const int* __restrict__ ei,
                                                float* __restrict__ deg) {
  int e = blockIdx.x * blockDim.x + threadIdx.x;
  if (e < N_EDGES) {
    int d = ei[N_EDGES + e];           // edge_index[1][e]
    unsafeAtomicAdd(&deg[d], 1.0f);    // global_atomic_add_f32
  }
}

// ---------------- deg -> deg^-1/2 in place (deg >= 1 always; guard anyway)
__global__ void __launch_bounds__(256) k_rsqrt(float* __restrict__ deg) {
  int i = blockIdx.x * blockDim.x + threadIdx.x;
  if (i < N_NODES) {
    float d = deg[i];
    deg[i] = (d > 0.0f) ? rsqrtf(d) : 0.0f;
  }
}

// ---------------- GEMM1: H[N,32] = X[N,128] @ W1[128,32] via WMMA f32 16x16x4
// One wave per 16x16 tile. 6250 row-tiles x 2 col-tiles = 12500 waves.
__global__ void __launch_bounds__(128) k_gemm1(const float* __restrict__ X,
                                               const float* __restrict__ W,
                                               float* __restrict__ H) {
  const int wid    = (blockIdx.x * blockDim.x + threadIdx.x) >> 5;
  const int lane   = threadIdx.x & 31;
  const int tile_m = wid >> 1;            // 0..6249
  const int tile_n = wid & 1;             // 0..1
  const int half   = lane >> 4;           // 0 = lanes 0-15, 1 = lanes 16-31
  const int l15    = lane & 15;
  const int rowA   = tile_m * 16 + l15;   // A-matrix row for this lane
  const int kOff   = half * 2;            // this half loads K+0/1 or K+2/3
  const int colB   = tile_n * 16 + l15;   // B-matrix column for this lane

  v8f c = {};
#pragma unroll
  for (int k = 0; k < IN_DIM; k += 4) {
    // A 16x4: VGPR0 = K+kOff, VGPR1 = K+kOff+1 (contiguous -> 8B load)
    v2f a = *(const v2f*)(X + (size_t)rowA * IN_DIM + k + kOff);
    // B 4x16: rows K+kOff and K+kOff+1, column colB
    v2f b;
    b.x = W[(k + kOff + 0) * HID_DIM + colB];
    b.y = W[(k + kOff + 1) * HID_DIM + colB];
    c = __builtin_amdgcn_wmma_f32_16x16x4_f32(false, a, false, b,
                                              (short)0, c, false, false);
  }
  // C/D: VGPR i -> row m0 + half*8 + i, col = tile_n*16 + l15
  const int m0 = tile_m * 16;
#pragma unroll
  for (int i = 0; i < 8; ++i) {
    H[(size_t)(m0 + half * 8 + i) * HID_DIM + tile_n * 16 + l15] = c[i];
  }
}

// ---------------- layer-1 aggregation: one edge per wave, one feat per lane
__global__ void __launch_bounds__(256) k_agg32(const int* __restrict__ ei,
                                               const float* __restrict__ dinv,
                                               const float* __restrict__ h,
                                               float* __restrict__ agg) {
  int idx = blockIdx.x * blockDim.x + threadIdx.x;
  int e = idx >> 5;                  // edge id (incl. self loops)
  int f = idx & 31;                  // feature
  const int total = N_EDGES + N_NODES;
  if (e >= total) return;
  int s, d;
  if (e < N_EDGES) { s = ei[e]; d = ei[N_EDGES + e]; }
  else             { s = d = e - N_EDGES; }           // self loop
  float w = dinv[s] * dinv[d];
  float v = h[(size_t)s * HID_DIM + f] * w;
  unsafeAtomicAdd(&agg[(size_t)d * HID_DIM + f], v);
}

// ---------------- bias + ReLU in place
__global__ void __launch_bounds__(256) k_bias_relu(float* __restrict__ a,
                                                   const float* __restrict__ b1) {
  int i = blockIdx.x * blockDim.x + threadIdx.x;
  if (i < N_NODES * HID_DIM) {
    float v = a[i] + b1[i & (HID_DIM - 1)];
    a[i] = fmaxf(v, 0.0f);
  }
}

// ---------------- GEMM2: G[N,16] = H1[N,32] @ W2[32,16] via WMMA f32 16x16x4
// One wave per 16x16 tile, single column tile. 6250 waves.
__global__ void __launch_bounds__(64) k_gemm2(const float* __restrict__ H1,
                                              const float* __restrict__ W,
                                              float* __restrict__ G) {
  const int wid    = (blockIdx.x * blockDim.x + threadIdx.x) >> 5;
  const int lane   = threadIdx.x & 31;
  const int tile_m = wid;                 // 0..6249
  const int half   = lane >> 4;
  const int l15    = lane & 15;
  const int rowA   = tile_m * 16 + l15;
  const int kOff   = half * 2;
  const int colB   = l15;

  v8f c = {};
#pragma unroll
  for (int k = 0; k < HID_DIM; k += 4) {
    v2f a = *(const v2f*)(H1 + (size_t)rowA * HID_DIM + k + kOff);
    v2f b;
    b.x = W[(k + kOff + 0) * OUT_DIM + colB];
    b.y = W[(k + kOff + 1) * OUT_DIM + colB];
    c = __builtin_amdgcn_wmma_f32_16x16x4_f32(false, a, false, b,
                                              (short)0, c, false, false);
  }
  const int m0 = tile_m * 16;
#pragma unroll
  for (int i = 0; i < 8; ++i) {
    G[(size_t)(m0 + half * 8 + i) * OUT_DIM + l15] = c[i];
  }
}

// ---------------- layer-2 aggregation: 2 edges per wave, 16 feats per edge
__global__ void __launch_bounds__(256) k_agg16(const int* __restrict__ ei,
                                               const float* __restrict__ dinv,
                                               const float* __restrict__ g,
                                               float* __restrict__ out) {
  int idx = blockIdx.x * blockDim.x + threadIdx.x;
  int e = idx >> 4;
  int f = idx & 15;
  const int total = N_EDGES + N_NODES;
  if (e >= total) return;
  int s, d;
  if (e < N_EDGES) { s = ei[e]; d = ei[N_EDGES + e]; }
  else             { s = d = e - N_EDGES; }
  float w = dinv[s] * dinv[d];
  float v = g[(size_t)s * OUT_DIM + f] * w;
  unsafeAtomicAdd(&out[(size_t)d * OUT_DIM + f], v);   // out pre-seeded with b2
}

// ---------------------------------------------------------------------------
extern "C" void kernel_launch(void* const* d_in, const int* in_sizes, int n_in,
                              void* d_out, int out_size, void* d_ws, size_t ws_size,
                              hipStream_t stream) {
  const float* x  = (const float*)d_in[0];
  const float* W1 = (const float*)d_in[1];
  const float* b1 = (const float*)d_in[2];
  const float* W2 = (const float*)d_in[3];
  const float* b2 = (const float*)d_in[4];
  const int*   ei = (const int*)d_in[5];   // edge_index [2, E] (row-major)

  // workspace layout (fp32): deg/dinv[N] | h[N*32] | agg1/h1[N*32] | g[N*16]
  float* deg = (float*)d_ws;
  float* h   = deg + N_NODES;
  float* h1  = h   + (size_t)N_NODES * HID_DIM;
  float* g   = h1  + (size_t)N_NODES * HID_DIM;
  float* out = (float*)d_out;

  const int B = 256;
  // 1. init deg=1 (self loop), agg1=0, out=b2
  k_init<<<(N_NODES * HID_DIM + B - 1) / B, B, 0, stream>>>(deg, h1, out, b2);
  // 2. in-degree
  k_degree<<<(N_EDGES + B - 1) / B, B, 0, stream>>>(ei, deg);
  // 3. deg^-1/2
  k_rsqrt<<<(N_NODES + B - 1) / B, B, 0, stream>>>(deg);
  // 4. H = X @ W1   (12500 waves, 4 waves/block)
  k_gemm1<<<3125, 128, 0, stream>>>(x, W1, h);
  // 5. agg1[dst] += norm * H[src]
  {
    long long threads = (long long)(N_EDGES + N_NODES) * HID_DIM;
    k_agg32<<<(unsigned)((threads + B - 1) / B), B, 0, stream>>>(ei, deg, h, h1);
  }
  // 6. h1 = relu(agg1 + b1)
  k_bias_relu<<<(N_NODES * HID_DIM + B - 1) / B, B, 0, stream>>>(h1, b1);
  // 7. G = H1 @ W2   (6250 waves, 2 waves/block)
  k_gemm2<<<3125, 64, 0, stream>>>(h1, W2, g);
  // 8. out[dst] += norm * G[src]  (out already holds b2)
  {
    long long threads = (long long)(N_EDGES + N_NODES) * OUT_DIM;
    k_agg16<<<(unsigned)((threads + B - 1) / B), B, 0, stream>>>(ei, deg, g, out);
  }
}